// GAT_10247791969041
// MI455X (gfx1250) — compile-verified
//
#include <hip/hip_runtime.h>
#include <math.h>

// GAT forward: h = X@W (WMMA f32), per-head attention dots, edge-softmax
// via encoded-uint atomicMax + atomicAdd, weighted scatter aggregation.

#define HEADS     4
#define HC        128     // HEADS * C
#define NEG_SLOPE 0.2f

typedef __attribute__((ext_vector_type(2))) float v2f;
typedef __attribute__((ext_vector_type(8))) float v8f;

// ---------------------------------------------------------------------------
// Kernel 1: Hout[N,128] = X[N,K] @ W[K,128] using V_WMMA_F32_16X16X4_F32.
// One block = 256 threads = 8 waves; block b covers rows [16b,16b+16),
// wave w covers cols [16w,16w+16). 64 WMMA steps of K=4.
//
// VGPR layouts per CDNA5 ISA 7.12.2:
//  A (16x4):  lanes 0-15 -> M=lane, VGPR{0,1}=K{0,1}; lanes 16-31 -> K{2,3}
//  B (4x16):  VGPR r: lanes 0-15 -> K=r (N=lane); lanes 16-31 -> K=r+2
//  C/D:       VGPR r: lanes 0-15 -> M=r, N=lane; lanes 16-31 -> M=r+8
// ---------------------------------------------------------------------------
__global__ void gat_gemm_wmma(const float* __restrict__ X,
                              const float* __restrict__ W,
                              float* __restrict__ Hout,
                              int K) {
  const int m0   = blockIdx.x << 4;
  const int wave = threadIdx.x >> 5;
  const int lane = threadIdx.x & 31;
  const int n0   = wave << 4;
  const int half = lane >> 4;       // 0: lanes 0-15, 1: lanes 16-31
  const int l15  = lane & 15;
  const int row  = m0 + l15;        // A row this lane supplies
  const int col  = n0 + l15;        // B col this lane supplies

  v8f acc = {};
  #pragma unroll 4
  for (int k = 0; k < K; k += 4) {
    const int kk = k + (half << 1);           // K index pair {kk, kk+1}
    const float2 av = *(const float2*)(X + (size_t)row * K + kk);
    v2f a; a.x = av.x; a.y = av.y;
    v2f b;
    b.x = W[(size_t)kk * HC + col];
    b.y = W[(size_t)(kk + 1) * HC + col];
    acc = __builtin_amdgcn_wmma_f32_16x16x4_f32(
        /*neg_a=*/false, a, /*neg_b=*/false, b,
        /*c_mod=*/(short)0, acc, /*reuse_a=*/false, /*reuse_b=*/false);
  }
  #pragma unroll
  for (int r = 0; r < 8; ++r) {
    const int rr = m0 + r + (half << 3);
    Hout[(size_t)rr * HC + col] = acc[r];
  }
}

// ---------------------------------------------------------------------------
// Kernel 2: a_src[n,h] = <h[n,h,:], att_src[h,:]>, same for a_dst.
// Block = 128 threads = 4 waves; wave w == head w (wave32 => 32 lanes == C).
// ---------------------------------------------------------------------------
__global__ void gat_attn_dots(const float* __restrict__ Hout,
                              const float* __restrict__ att_src,
                              const float* __restrict__ att_dst,
                              float* __restrict__ a_src,
                              float* __restrict__ a_dst) {
  const int node = blockIdx.x;
  const int t    = threadIdx.x;       // 0..127 == hd*32 + c
  const float hv = Hout[(size_t)node * HC + t];
  float vs = hv * att_src[t];
  float vd = hv * att_dst[t];
  #pragma unroll
  for (int off = 16; off > 0; off >>= 1) {
    vs += __shfl_down(vs, off);
    vd += __shfl_down(vd, off);
  }
  if ((t & 31) == 0) {
    const int hd = t >> 5;
    a_src[node * HEADS + hd] = vs;
    a_dst[node * HEADS + hd] = vd;
  }
}

// ---------------------------------------------------------------------------
// Monotonic order-preserving float<->uint encoding for atomicMax on floats.
// ---------------------------------------------------------------------------
__device__ __forceinline__ unsigned fenc(float f) {
  unsigned u = __float_as_uint(f);
  return (u & 0x80000000u) ? ~u : (u | 0x80000000u);
}
__device__ __forceinline__ float fdec(unsigned u) {
  u = (u & 0x80000000u) ? (u & 0x7FFFFFFFu) : ~u;
  return __uint_as_float(u);
}

// out[n,j] = bias[j]; segmax = enc(-inf); denom = 0
__global__ void gat_init(float* __restrict__ out, const float* __restrict__ bias,
                         unsigned* __restrict__ segmax_u, float* __restrict__ denom,
                         long long totalOut, long long totalNH) {
  const long long i = (long long)blockIdx.x * blockDim.x + threadIdx.x;
  if (i < totalOut) out[i] = bias[i & (HC - 1)];
  if (i < totalNH) { segmax_u[i] = 0x007FFFFFu; denom[i] = 0.0f; }  // enc(-inf)
}

__device__ __forceinline__ void edge_sd(const int* __restrict__ src,
                                        const int* __restrict__ dst,
                                        int eid, int E, int& s, int& d) {
  if (eid < E) { s = src[eid]; d = dst[eid]; }
  else         { s = d = eid - E; }            // self loops appended
}

__device__ __forceinline__ float lrelu(float e) {
  return e > 0.0f ? e : NEG_SLOPE * e;
}

// Pass A: per (edge, head) -> leaky_relu score -> segment max over dst.
__global__ void gat_edge_max(const int* __restrict__ src, const int* __restrict__ dst,
                             const float* __restrict__ a_src, const float* __restrict__ a_dst,
                             unsigned* __restrict__ segmax_u, int E, int Et) {
  const long long idx = (long long)blockIdx.x * blockDim.x + threadIdx.x;
  if (idx >= (long long)Et * HEADS) return;
  const int eid = (int)(idx >> 2);
  const int hd  = (int)(idx & 3);
  int s, d; edge_sd(src, dst, eid, E, s, d);
  const float e = lrelu(a_src[s * HEADS + hd] + a_dst[d * HEADS + hd]);
  atomicMax(&segmax_u[d * HEADS + hd], fenc(e));
}

// Fixup: decode, apply reference's "isfinite ? m : 0".
__global__ void gat_segmax_fix(const unsigned* __restrict__ segmax_u,
                               float* __restrict__ segmax_f, int n) {
  const int i = blockIdx.x * blockDim.x + threadIdx.x;
  if (i >= n) return;
  const float m = fdec(segmax_u[i]);
  segmax_f[i] = isfinite(m) ? m : 0.0f;
}

// Pass B: denom[dst,h] += exp(e - segmax[dst,h])
__global__ void gat_edge_denom(const int* __restrict__ src, const int* __restrict__ dst,
                               const float* __restrict__ a_src, const float* __restrict__ a_dst,
                               const float* __restrict__ segmax_f, float* __restrict__ denom,
                               int E, int Et) {
  const long long idx = (long long)blockIdx.x * blockDim.x + threadIdx.x;
  if (idx >= (long long)Et * HEADS) return;
  const int eid = (int)(idx >> 2);
  const int hd  = (int)(idx & 3);
  int s, d; edge_sd(src, dst, eid, E, s, d);
  const float e  = lrelu(a_src[s * HEADS + hd] + a_dst[d * HEADS + hd]);
  const float ex = __expf(e - segmax_f[d * HEADS + hd]);
  atomicAdd(&denom[d * HEADS + hd], ex);
}

// Pass C: one wave per edge; lane owns 4 channels of the 128.
// out[dst,h,c] += alpha * h[src,h,c]   (alpha recomputed per lane, L2-hot)
__global__ void gat_edge_scatter(const int* __restrict__ src, const int* __restrict__ dst,
                                 const float* __restrict__ a_src, const float* __restrict__ a_dst,
                                 const float* __restrict__ segmax_f, const float* __restrict__ denom,
                                 const float* __restrict__ Hout, float* __restrict__ out,
                                 int E, int Et) {
  const long long widx = ((long long)blockIdx.x * blockDim.x + threadIdx.x) >> 5;
  if (widx >= Et) return;
  const int lane = threadIdx.x & 31;
  const int eid  = (int)widx;
  int s, d; edge_sd(src, dst, eid, E, s, d);
  const int hd = lane >> 3;                  // 4 channels/lane -> head = lane/8
  const float e     = lrelu(a_src[s * HEADS + hd] + a_dst[d * HEADS + hd]);
  const float ex    = __expf(e - segmax_f[d * HEADS + hd]);
  const float alpha = ex / (denom[d * HEADS + hd] + 1e-16f);

  const int ch = lane << 2;                  // first of this lane's 4 channels
  const float4 hv = *(const float4*)(Hout + (size_t)s * HC + ch);
  float* op = out + (size_t)d * HC + ch;
  atomicAdd(op + 0, hv.x * alpha);
  atomicAdd(op + 1, hv.y * alpha);
  atomicAdd(op + 2, hv.z * alpha);
  atomicAdd(op + 3, hv.w * alpha);
}

// ---------------------------------------------------------------------------
extern "C" void kernel_launch(void* const* d_in, const int* in_sizes, int n_in,
                              void* d_out, int out_size, void* d_ws, size_t ws_size,
                              hipStream_t stream) {
  const float* X        = (const float*)d_in[0];   // [N, K]
  const int*   edges    = (const int*)  d_in[1];   // [2, E]
  const float* W        = (const float*)d_in[2];   // [K, 128]
  const float* att_src  = (const float*)d_in[3];   // [4, 32]
  const float* att_dst  = (const float*)d_in[4];   // [4, 32]
  const float* bias     = (const float*)d_in[5];   // [128]
  float*       out      = (float*)d_out;           // [N, 128]

  const int K  = in_sizes[2] / HC;                 // 256
  const int Nn = in_sizes[0] / K;                  // 100000 (divisible by 16)
  const int E  = in_sizes[1] / 2;                  // 1600000
  const int Et = E + Nn;                           // + self loops
  const int* src = edges;
  const int* dst = edges + E;

  // Workspace carve-up (~59 MB)
  char* ws = (char*)d_ws;
  float*    Hout     = (float*)ws;    ws += (size_t)Nn * HC    * sizeof(float);
  float*    a_src    = (float*)ws;    ws += (size_t)Nn * HEADS * sizeof(float);
  float*    a_dst    = (float*)ws;    ws += (size_t)Nn * HEADS * sizeof(float);
  unsigned* segmax_u = (unsigned*)ws; ws += (size_t)Nn * HEADS * sizeof(unsigned);
  float*    segmax_f = (float*)ws;    ws += (size_t)Nn * HEADS * sizeof(float);
  float*    denom    = (float*)ws;    ws += (size_t)Nn * HEADS * sizeof(float);

  const long long totalOut = (long long)Nn * HC;
  const long long totalNH  = (long long)Nn * HEADS;
  const long long totalEH  = (long long)Et * HEADS;

  // 0. init: out = bias rows, segmax = enc(-inf), denom = 0
  gat_init<<<(int)((totalOut + 255) / 256), 256, 0, stream>>>(
      out, bias, segmax_u, denom, totalOut, totalNH);

  // 1. h = X @ W  (WMMA f32 16x16x4)
  gat_gemm_wmma<<<(Nn + 15) / 16, 256, 0, stream>>>(X, W, Hout, K);

  // 2. per-node attention dots
  gat_attn_dots<<<Nn, HC, 0, stream>>>(Hout, att_src, att_dst, a_src, a_dst);

  // 3. segment max over destination neighborhoods
  gat_edge_max<<<(int)((totalEH + 255) / 256), 256, 0, stream>>>(
      src, dst, a_src, a_dst, segmax_u, E, Et);
  gat_segmax_fix<<<(int)((totalNH + 255) / 256), 256, 0, stream>>>(
      segmax_u, segmax_f, (int)totalNH);

  // 4. softmax denominator
  gat_edge_denom<<<(int)((totalEH + 255) / 256), 256, 0, stream>>>(
      src, dst, a_src, a_dst, segmax_f, denom, E, Et);

  // 5. weighted scatter aggregation (one wave per edge)
  gat_edge_scatter<<<(Et + 7) / 8, 256, 0, stream>>>(
      src, dst, a_src, a_dst, segmax_f, denom, Hout, out, E, Et);
}